// SkipGramNEGModel_75153337745589
// MI455X (gfx1250) — compile-verified
//
#include <hip/hip_runtime.h>
#include <math.h>

// SkipGram negative-sampling loss for MI455X (gfx1250, wave32).
// Memory-bound gather workload (~224 MB reads, ~0.1 GFLOP); tables (102 MB)
// fit in the 192 MB L2. Dot products are batched 16 rows at a time through
// V_WMMA_F32_16X16X4_F32 (exact fp32 math); only the diagonal of each 16x16
// product is needed, which is free at this arithmetic intensity.

typedef float v2f __attribute__((ext_vector_type(2)));
typedef float v8f __attribute__((ext_vector_type(8)));

#define EMBED 128
#define NEG 5
#define WAVES_PER_BLOCK 8
#define ROWS_PER_WAVE 16

__device__ __forceinline__ float diag_extract(v8f c, int idx) {
    // Select c[idx] without dynamic indexing (keeps everything in VGPRs).
    float r = c[0];
    r = (idx == 1) ? c[1] : r;
    r = (idx == 2) ? c[2] : r;
    r = (idx == 3) ? c[3] : r;
    r = (idx == 4) ? c[4] : r;
    r = (idx == 5) ? c[5] : r;
    r = (idx == 6) ? c[6] : r;
    r = (idx == 7) ? c[7] : r;
    return r;
}

__device__ __forceinline__ float clip10(float x) {
    return fminf(fmaxf(x, -10.0f), 10.0f);
}

__device__ __forceinline__ float log_sigmoid(float x) {
    // log(sigmoid(x)) = min(x,0) - log1p(exp(-|x|)); x is pre-clipped to [-10,10]
    return fminf(x, 0.0f) - log1pf(expf(-fabsf(x)));
}

__global__ __launch_bounds__(256) void skipgram_neg_wmma_kernel(
    const int* __restrict__ pos_w,
    const int* __restrict__ pos_v,
    const int* __restrict__ neg_v,
    const float* __restrict__ w_table,
    const float* __restrict__ v_table,
    float* __restrict__ partials) {
    const int lane = threadIdx.x & 31;
    const int wave = threadIdx.x >> 5;
    const int tile = blockIdx.x * WAVES_PER_BLOCK + wave;   // 16 batch rows per wave

    const int m    = lane & 15;   // row within tile (A: M index, B: N index)
    const int half = lane >> 4;   // K-group: lanes 0-15 -> K{0,1}, lanes 16-31 -> K{2,3}
    const int i    = tile * ROWS_PER_WAVE + m;              // batch row this lane gathers

    // Gather row indices (one w row + 6 v rows per batch row).
    const int wrow = pos_w[i];
    const int vr0  = pos_v[i];
    const int vr1  = neg_v[i * NEG + 0];
    const int vr2  = neg_v[i * NEG + 1];
    const int vr3  = neg_v[i * NEG + 2];
    const int vr4  = neg_v[i * NEG + 3];
    const int vr5  = neg_v[i * NEG + 4];

    // Per-lane base pointers: K offset = 2*half + j within each 4-wide chunk.
    const float* pw  = w_table + (size_t)wrow * EMBED + half * 2;
    const float* pv0 = v_table + (size_t)vr0  * EMBED + half * 2;
    const float* pv1 = v_table + (size_t)vr1  * EMBED + half * 2;
    const float* pv2 = v_table + (size_t)vr2  * EMBED + half * 2;
    const float* pv3 = v_table + (size_t)vr3  * EMBED + half * 2;
    const float* pv4 = v_table + (size_t)vr4  * EMBED + half * 2;
    const float* pv5 = v_table + (size_t)vr5  * EMBED + half * 2;

    v8f c0 = {}, c1 = {}, c2 = {}, c3 = {}, c4 = {}, c5 = {};

    // 32 K-chunks of 4; A chunk loaded once, reused across the 6 B matrices.
    // EXEC is all-ones for every WMMA (no divergence up to here).
#pragma unroll 4
    for (int kc = 0; kc < EMBED / 4; ++kc) {
        const int ko = kc * 4;
        v2f a = *(const v2f*)(pw + ko);
        v2f b;
        b = *(const v2f*)(pv0 + ko);
        c0 = __builtin_amdgcn_wmma_f32_16x16x4_f32(false, a, false, b, (short)0, c0, false, false);
        b = *(const v2f*)(pv1 + ko);
        c1 = __builtin_amdgcn_wmma_f32_16x16x4_f32(false, a, false, b, (short)0, c1, false, false);
        b = *(const v2f*)(pv2 + ko);
        c2 = __builtin_amdgcn_wmma_f32_16x16x4_f32(false, a, false, b, (short)0, c2, false, false);
        b = *(const v2f*)(pv3 + ko);
        c3 = __builtin_amdgcn_wmma_f32_16x16x4_f32(false, a, false, b, (short)0, c3, false, false);
        b = *(const v2f*)(pv4 + ko);
        c4 = __builtin_amdgcn_wmma_f32_16x16x4_f32(false, a, false, b, (short)0, c4, false, false);
        b = *(const v2f*)(pv5 + ko);
        c5 = __builtin_amdgcn_wmma_f32_16x16x4_f32(false, a, false, b, (short)0, c5, false, false);
    }

    // Diagonal D[r][r]: rows 0-7 -> lane r, VGPR r; rows 8-15 -> lane r+16, VGPR r-8.
    const bool has_diag = (lane < 8) || (lane >= 24);
    const int idx = (lane < 8) ? lane : (lane - 24);

    float s = clip10(diag_extract(c0, idx));
    float contrib = log_sigmoid(s);                 // positive score
    s = clip10(diag_extract(c1, idx)); contrib += log_sigmoid(-s);
    s = clip10(diag_extract(c2, idx)); contrib += log_sigmoid(-s);
    s = clip10(diag_extract(c3, idx)); contrib += log_sigmoid(-s);
    s = clip10(diag_extract(c4, idx)); contrib += log_sigmoid(-s);
    s = clip10(diag_extract(c5, idx)); contrib += log_sigmoid(-s);
    if (!has_diag) contrib = 0.0f;

    // Wave32 butterfly reduction.
#pragma unroll
    for (int off = 16; off > 0; off >>= 1)
        contrib += __shfl_xor(contrib, off, 32);

    __shared__ float wsum[WAVES_PER_BLOCK];
    if (lane == 0) wsum[wave] = contrib;
    __syncthreads();
    if (threadIdx.x == 0) {
        float t = 0.0f;
#pragma unroll
        for (int j = 0; j < WAVES_PER_BLOCK; ++j) t += wsum[j];
        partials[blockIdx.x] = t;
    }
}

__global__ __launch_bounds__(256) void skipgram_reduce_kernel(
    const float* __restrict__ partials, int n, float* __restrict__ out) {
    __shared__ float buf[256];
    float t = 0.0f;
    for (int j = threadIdx.x; j < n; j += 256) t += partials[j];
    buf[threadIdx.x] = t;
    __syncthreads();
#pragma unroll
    for (int s = 128; s > 0; s >>= 1) {
        if (threadIdx.x < s) buf[threadIdx.x] += buf[threadIdx.x + s];
        __syncthreads();
    }
    if (threadIdx.x == 0) out[0] = -buf[0];   // loss = -(sum of log-sigmoids)
}

extern "C" void kernel_launch(void* const* d_in, const int* in_sizes, int n_in,
                              void* d_out, int out_size, void* d_ws, size_t ws_size,
                              hipStream_t stream) {
    const int*   pos_w   = (const int*)d_in[0];
    const int*   pos_v   = (const int*)d_in[1];
    const int*   neg_v   = (const int*)d_in[2];
    const float* w_table = (const float*)d_in[3];
    const float* v_table = (const float*)d_in[4];
    float* out = (float*)d_out;

    const int batch  = in_sizes[0];                          // 65536
    const int tiles  = batch / ROWS_PER_WAVE;                // 4096 waves
    const int blocks = tiles / WAVES_PER_BLOCK;              // 512 blocks of 8 waves

    float* partials = (float*)d_ws;                          // 512 floats of scratch

    skipgram_neg_wmma_kernel<<<blocks, 256, 0, stream>>>(
        pos_w, pos_v, neg_v, w_table, v_table, partials);
    skipgram_reduce_kernel<<<1, 256, 0, stream>>>(partials, blocks, out);
}